// FilterNet24H2_89318139888367
// MI455X (gfx1250) — compile-verified
//
#include <hip/hip_runtime.h>

// ---------------------------------------------------------------------------
// FilterNet forward for MI455X (gfx1250, wave32).
// GEMMs run on V_WMMA_F32_16X16X4_F32 (fp32 matrix pipe, exact-precision path).
// ---------------------------------------------------------------------------

typedef __attribute__((ext_vector_type(2))) float v2f;
typedef __attribute__((ext_vector_type(8))) float v8f;

#define BATCH   65536
#define LWIN    240
#define NIN     48      // 32 emb + 16 cont
#define H1N     512
#define H2N     256
#define OUTN    24
#define KTOT    364     // 108 wave feats + 256 h2

// LDS strides (floats). Even (keeps 8B alignment for ds_load_b64) and
// stride mod 64 == 2*odd -> 16 lanes map to distinct banks.
#define X_STRIDE   50
#define XF_STRIDE  366
#define H1_STRIDE  522
#define XW_STRIDE  242

struct ConvW { const float* w[12]; };   // wk1a,wk1b,...,wk6a,wk6b

__device__ __forceinline__ v8f wmma_f32(v2f a, v2f b, v8f c) {
  // D(16x16,f32) = A(16x4,f32) * B(4x16,f32) + C
  return __builtin_amdgcn_wmma_f32_16x16x4_f32(
      /*neg_a=*/false, a, /*neg_b=*/false, b,
      /*c_mod=*/(short)0, c, /*reuse_a=*/false, /*reuse_b=*/false);
}

__device__ __forceinline__ v8f zero8() {
  v8f z = {0.f, 0.f, 0.f, 0.f, 0.f, 0.f, 0.f, 0.f};
  return z;
}

// ---------------------------------------------------------------------------
// Conv-branch features, templated on kernel size so pooling segment bounds
// constant-fold and fa[] stays in registers.
// fa = leaky(conv1(xw,wa,1) @ P(lin,12)), fb = leaky(conv1(fa,wb,2) @ P(lin2,6))
// ---------------------------------------------------------------------------
template <int KS>
__device__ __forceinline__ void branch_feats(const float* __restrict__ xwrow,
                                             const float* __restrict__ wag,
                                             const float* __restrict__ wbg,
                                             float* __restrict__ outp) {
  constexpr int LIN = LWIN - (KS - 1);
  float wa[KS], wb[KS];
#pragma unroll
  for (int k = 0; k < KS; ++k) { wa[k] = wag[k]; wb[k] = wbg[k]; }

  float fa[12];
#pragma unroll
  for (int i = 0; i < 12; ++i) {
    const int s = (i * LIN) / 12;
    const int e = ((i + 1) * LIN + 11) / 12;   // ceil
    float acc = 0.f;
    for (int j = s; j < e; ++j) {
      float c = 0.f;
#pragma unroll
      for (int k = 0; k < KS; ++k) c = __builtin_fmaf(wa[k], xwrow[j + k], c);
      acc += c;
    }
    float v = acc * (1.0f / (float)(e - s));
    v = (v > 0.f) ? v : 0.1f * v;               // leaky_relu 0.1
    fa[i] = v;
    outp[i] = v;
  }

  constexpr int LIN2 = 12 - 2 * (KS - 1);
#pragma unroll
  for (int i = 0; i < 6; ++i) {
    const int s = (i * LIN2) / 6;
    const int e = ((i + 1) * LIN2 + 5) / 6;     // ceil
    float acc = 0.f;
#pragma unroll
    for (int j = s; j < e; ++j) {
      float c = 0.f;
#pragma unroll
      for (int k = 0; k < KS; ++k) c = __builtin_fmaf(wb[k], fa[j + 2 * k], c);
      acc += c;
    }
    float v = acc * (1.0f / (float)(e - s));
    outp[12 + i] = (v > 0.f) ? v : 0.1f * v;
  }
}

// ---------------------------------------------------------------------------
// Batch-norm statistics: deterministic two-stage tree reduction (no float
// atomics, so timed replays are bit-identical).
// ---------------------------------------------------------------------------
__global__ __launch_bounds__(256) void bn_partial_kernel(
    const float* __restrict__ x_cont, float* __restrict__ part) {
  __shared__ float s1[256], s2[256];
  const int t = threadIdx.x, b = blockIdx.x;
  const float* p = x_cont + (size_t)b * 4096;   // 256 rows * 16 cols
  float a1 = 0.f, a2 = 0.f;
#pragma unroll
  for (int k = 0; k < 16; ++k) {                // idx%16 == t%16 always
    float v = p[t + k * 256];
    a1 += v;
    a2 = __builtin_fmaf(v, v, a2);
  }
  s1[t] = a1; s2[t] = a2;
  __syncthreads();
  if (t < 16) {
    float r1 = 0.f, r2 = 0.f;
#pragma unroll
    for (int j = 0; j < 16; ++j) { r1 += s1[t + 16 * j]; r2 += s2[t + 16 * j]; }
    part[b * 32 + t]      = r1;
    part[b * 32 + 16 + t] = r2;
  }
}

__global__ __launch_bounds__(32) void bn_final_kernel(
    const float* __restrict__ part, const float* __restrict__ bn_g,
    const float* __restrict__ bn_b, float* __restrict__ scsh) {
  const int t = threadIdx.x;
  if (t < 16) {
    float s1 = 0.f, s2 = 0.f;
    for (int j = 0; j < 256; ++j) { s1 += part[j * 32 + t]; s2 += part[j * 32 + 16 + t]; }
    const float m   = s1 * (1.0f / (float)BATCH);
    const float var = s2 * (1.0f / (float)BATCH) - m * m;
    const float sc  = bn_g[t] * rsqrtf(var + 1e-5f);
    scsh[t]      = sc;                 // scale
    scsh[16 + t] = bn_b[t] - m * sc;   // shift
  }
}

// ---------------------------------------------------------------------------
// Fused main kernel: one block (8 wave32) per 16-row tile.
// ---------------------------------------------------------------------------
__global__ __launch_bounds__(256) void filternet_main(
    const float* __restrict__ x_window, const int* __restrict__ x_cat,
    const float* __restrict__ x_cont, ConvW cw,
    const float* __restrict__ E0, const float* __restrict__ E1,
    const float* __restrict__ E2,
    const float* __restrict__ W1, const float* __restrict__ b1,
    const float* __restrict__ W2, const float* __restrict__ b2,
    const float* __restrict__ Wlin,
    const float* __restrict__ bn_scale, const float* __restrict__ bn_shift,
    float* __restrict__ out) {
  __shared__ float s_xf[16 * XF_STRIDE];   // [ffc(108) | h2(256)] per row
  __shared__ float s_x [16 * X_STRIDE];    // MLP input 16x48
  __shared__ float s_big[16 * H1_STRIDE];  // aliased: xw (16x240) then h1 (16x512)

  const int t    = threadIdx.x;
  const int row0 = blockIdx.x * 16;
  const int lane = t & 31;
  const int wv   = t >> 5;                 // wave id 0..7
  const int half = (lane >> 4) & 1;        // K-half selector for WMMA frags
  const int l16  = lane & 15;

  // ---- stage 0: stage window rows (coalesced) + build MLP input ----
  for (int idx = t; idx < 16 * LWIN; idx += 256) {
    int r = idx / LWIN, c = idx - r * LWIN;
    s_big[r * XW_STRIDE + c] = x_window[(size_t)row0 * LWIN + idx];
  }
  for (int idx = t; idx < 16 * NIN; idx += 256) {
    int r = idx / NIN, c = idx - r * NIN;
    int grow = row0 + r;
    float v;
    if (c < 20)      v = E0[x_cat[grow * 3 + 0] * 20 + c];
    else if (c < 24) v = E1[x_cat[grow * 3 + 1] * 4 + (c - 20)];
    else if (c < 32) v = E2[x_cat[grow * 3 + 2] * 8 + (c - 24)];
    else {
      int cc = c - 32;
      v = __builtin_fmaf(x_cont[(size_t)grow * 16 + cc], bn_scale[cc], bn_shift[cc]);
    }
    s_x[r * X_STRIDE + c] = v;
  }
  __syncthreads();

  // ---- stage 1: conv-branch features (16 rows x 6 branches on 96 threads) ----
  if (t < 96) {
    int r  = t & 15;
    int br = t >> 4;
    const float* xwrow = &s_big[r * XW_STRIDE];
    float* op = &s_xf[r * XF_STRIDE + 18 * br];
    switch (br) {
      case 0:  branch_feats<1>(xwrow, cw.w[0],  cw.w[1],  op); break;
      case 1:  branch_feats<2>(xwrow, cw.w[2],  cw.w[3],  op); break;
      case 2:  branch_feats<3>(xwrow, cw.w[4],  cw.w[5],  op); break;
      case 3:  branch_feats<4>(xwrow, cw.w[6],  cw.w[7],  op); break;
      case 4:  branch_feats<5>(xwrow, cw.w[8],  cw.w[9],  op); break;
      default: branch_feats<6>(xwrow, cw.w[10], cw.w[11], op); break;
    }
  }
  __syncthreads();   // xw reads done; s_big may now be overwritten with h1

  // ---- stage 2: h1 = relu(x @ W1 + b1); wave wv owns N-tiles wv*4..wv*4+3 ----
  v8f acc1[4];
#pragma unroll
  for (int i = 0; i < 4; ++i) acc1[i] = zero8();
  for (int k0 = 0; k0 < NIN; k0 += 4) {
    const int ka = k0 + 2 * half;
    v2f a;
    a.x = s_x[l16 * X_STRIDE + ka];
    a.y = s_x[l16 * X_STRIDE + ka + 1];
#pragma unroll
    for (int i = 0; i < 4; ++i) {
      const int n = (wv * 4 + i) * 16 + l16;
      v2f b;
      b.x = W1[ka * H1N + n];            // L2-resident weight stream
      b.y = W1[(ka + 1) * H1N + n];
      acc1[i] = wmma_f32(a, b, acc1[i]);
    }
  }
#pragma unroll
  for (int i = 0; i < 4; ++i) {
    const int n = (wv * 4 + i) * 16 + l16;
    const float bias = b1[n];
#pragma unroll
    for (int r = 0; r < 8; ++r) {
      float v = acc1[i][r] + bias;
      v = v > 0.f ? v : 0.f;
      s_big[(r + 8 * half) * H1_STRIDE + n] = v;   // C frag: row r+8*half, col n
    }
  }
  __syncthreads();

  // ---- stage 3: h2 = relu(h1 @ W2 + b2); wave wv owns N-tiles wv*2, wv*2+1 ----
  v8f acc2[2];
  acc2[0] = zero8(); acc2[1] = zero8();
#pragma unroll 2
  for (int k0 = 0; k0 < H1N; k0 += 4) {
    const int ka = k0 + 2 * half;
    v2f a;
    a.x = s_big[l16 * H1_STRIDE + ka];   // ds_load_b64
    a.y = s_big[l16 * H1_STRIDE + ka + 1];
#pragma unroll
    for (int i = 0; i < 2; ++i) {
      const int n = (wv * 2 + i) * 16 + l16;
      v2f b;
      b.x = W2[ka * H2N + n];
      b.y = W2[(ka + 1) * H2N + n];
      acc2[i] = wmma_f32(a, b, acc2[i]);
    }
  }
#pragma unroll
  for (int i = 0; i < 2; ++i) {
    const int n = (wv * 2 + i) * 16 + l16;
    const float bias = b2[n];
#pragma unroll
    for (int r = 0; r < 8; ++r) {
      float v = acc2[i][r] + bias;
      v = v > 0.f ? v : 0.f;
      s_xf[(r + 8 * half) * XF_STRIDE + 108 + n] = v;
    }
  }
  __syncthreads();

  // ---- stage 4: out = [ffc | h2] @ Wlin (K=364, N=24 -> waves 0,1) ----
  if (wv < 2) {
    v8f acc3 = zero8();
    const int n = wv * 16 + l16;
    const bool valid = (n < OUTN);
#pragma unroll 2
    for (int k0 = 0; k0 < KTOT; k0 += 4) {
      const int ka = k0 + 2 * half;
      v2f a;
      a.x = s_xf[l16 * XF_STRIDE + ka];
      a.y = s_xf[l16 * XF_STRIDE + ka + 1];
      v2f b;
      b.x = valid ? Wlin[ka * OUTN + n] : 0.f;       // pad cols 24..31 with 0
      b.y = valid ? Wlin[(ka + 1) * OUTN + n] : 0.f;
      acc3 = wmma_f32(a, b, acc3);
    }
    if (valid) {
#pragma unroll
      for (int r = 0; r < 8; ++r)
        out[(size_t)(row0 + r + 8 * half) * OUTN + n] = acc3[r];
    }
  }
}

// ---------------------------------------------------------------------------
extern "C" void kernel_launch(void* const* d_in, const int* in_sizes, int n_in,
                              void* d_out, int out_size, void* d_ws, size_t ws_size,
                              hipStream_t stream) {
  (void)in_sizes; (void)n_in; (void)out_size; (void)ws_size;
  const float* x_window = (const float*)d_in[0];
  const int*   x_cat    = (const int*)d_in[1];     // JAX x64 off -> int32
  const float* x_cont   = (const float*)d_in[2];
  ConvW cw;
  for (int i = 0; i < 12; ++i) cw.w[i] = (const float*)d_in[3 + i];
  const float* E0   = (const float*)d_in[15];
  const float* E1   = (const float*)d_in[16];
  const float* E2   = (const float*)d_in[17];
  const float* bn_g = (const float*)d_in[18];
  const float* bn_b = (const float*)d_in[19];
  const float* W1   = (const float*)d_in[20];
  const float* b1   = (const float*)d_in[21];
  const float* W2   = (const float*)d_in[22];
  const float* b2   = (const float*)d_in[23];
  const float* Wlin = (const float*)d_in[24];

  float* ws   = (float*)d_ws;
  float* part = ws;          // 256 blocks * 32 floats = 8192
  float* scsh = ws + 8192;   // 16 scale + 16 shift

  bn_partial_kernel<<<256, 256, 0, stream>>>(x_cont, part);
  bn_final_kernel<<<1, 32, 0, stream>>>(part, bn_g, bn_b, scsh);
  filternet_main<<<BATCH / 16, 256, 0, stream>>>(
      x_window, x_cat, x_cont, cw, E0, E1, E2,
      W1, b1, W2, b2, Wlin, scsh, scsh + 16, (float*)d_out);
}